// MacGCNBlock_25640954757830
// MI455X (gfx1250) — compile-verified
//
#include <hip/hip_runtime.h>
#include <hip/hip_bf16.h>
#include <stdint.h>

// Problem constants (match reference)
#define G_ 2
#define E_ 1000000
#define W_ 50000
#define S_ 100000
#define D_ 64
#define N_ 150000   // W + S

typedef __attribute__((ext_vector_type(2))) float v2f;
typedef __attribute__((ext_vector_type(8))) float v8f;

// ---------------- degree / normalization ----------------

__global__ void zero_deg_kernel(float* __restrict__ deg) {
  int i = blockIdx.x * blockDim.x + threadIdx.x;
  if (i < G_ * N_) deg[i] = 0.0f;
}

__global__ void degree_kernel(const int* __restrict__ rows,
                              const int* __restrict__ cols,
                              const float* __restrict__ vals,
                              float* __restrict__ deg) {
  int i = blockIdx.x * blockDim.x + threadIdx.x;
  if (i >= G_ * E_) return;
  int g = i / E_;
  float v = vals[i];
  unsafeAtomicAdd(deg + (size_t)g * N_ + rows[i], v);
  unsafeAtomicAdd(deg + (size_t)g * N_ + W_ + cols[i], v);
}

__global__ void invsqrt_kernel(float* __restrict__ deg) {
  int i = blockIdx.x * blockDim.x + threadIdx.x;
  if (i < G_ * N_) deg[i] = 1.0f / (sqrtf(deg[i]) + 1e-8f);
}

__global__ void edgeweight_kernel(const int* __restrict__ rows,
                                  const int* __restrict__ cols,
                                  const float* __restrict__ vals,
                                  const float* __restrict__ invs,
                                  float* __restrict__ vn) {
  int i = blockIdx.x * blockDim.x + threadIdx.x;
  if (i >= G_ * E_) return;
  int g = i / E_;
  const float* isg = invs + (size_t)g * N_;
  vn[i] = vals[i] * isg[rows[i]] * isg[W_ + cols[i]];
}

// ---------------- feature init: f0 = [wf;sf], acc(d_out) = f0, f1 = 0 ----------------

__global__ void init_feat_kernel(const float* __restrict__ wf,
                                 const float* __restrict__ sf,
                                 float* __restrict__ f0,
                                 float* __restrict__ f1,
                                 float* __restrict__ out) {
  size_t i = (size_t)blockIdx.x * blockDim.x + threadIdx.x; // one float4 each
  size_t fi = i * 4;                 // flat index into [G, N, D]
  size_t row = fi / D_;              // g*N + n
  int g = (int)(row / N_);
  int n = (int)(row % N_);
  int d = (int)(fi % D_);
  float4 v;
  size_t oidx;
  if (n < W_) {
    v = *(const float4*)(wf + ((size_t)g * W_ + n) * D_ + d);
    oidx = ((size_t)g * W_ + n) * D_ + d;
  } else {
    v = *(const float4*)(sf + ((size_t)g * S_ + (n - W_)) * D_ + d);
    oidx = (size_t)G_ * W_ * D_ + ((size_t)g * S_ + (n - W_)) * D_ + d;
  }
  *(float4*)(f0 + fi) = v;
  *(float4*)(out + oidx) = v;
  *(float4*)(f1 + fi) = make_float4(0.f, 0.f, 0.f, 0.f);
}

// ---------------- SpMM: ftgt += A_norm * fcur (symmetric edge pairs) ----------------
// 64 edges per block; edge streams staged into LDS with CDNA5 async-to-LDS loads.
// 4 threads per edge (16 floats each); scatter via fp32 global atomics (L2-resident).

#define CHUNK 64

__global__ __launch_bounds__(256) void spmm_kernel(const int* __restrict__ rows,
                                                   const int* __restrict__ cols,
                                                   const float* __restrict__ vn,
                                                   const float* __restrict__ fcur,
                                                   float* __restrict__ ftgt) {
  __shared__ int   s_rows[CHUNK];
  __shared__ int   s_cols[CHUNK];
  __shared__ float s_vn[CHUNK];

  const int t = threadIdx.x;
  const size_t base = (size_t)blockIdx.x * CHUNK;   // edge id in [0, G*E); E % CHUNK == 0
  const int g = (int)(base / E_);

  // Stage edge chunk into LDS via async-to-LDS (ASYNCcnt path). Waves 0..5 fully active.
  if (t < 3 * CHUNK) {
    const int a = t >> 6;      // 0: rows, 1: cols, 2: vn
    const int i = t & (CHUNK - 1);
    const void* gp = (a == 0) ? (const void*)(rows + base + i)
                   : (a == 1) ? (const void*)(cols + base + i)
                              : (const void*)(vn + base + i);
    uint32_t lds_off = (a == 0) ? (uint32_t)(uintptr_t)&s_rows[i]
                     : (a == 1) ? (uint32_t)(uintptr_t)&s_cols[i]
                                : (uint32_t)(uintptr_t)&s_vn[i];
    uint64_t gaddr = (uint64_t)(uintptr_t)gp;
    asm volatile("global_load_async_to_lds_b32 %0, %1, off"
                 :: "v"(lds_off), "v"(gaddr) : "memory");
  }
  asm volatile("s_wait_asynccnt 0x0" ::: "memory");
  __syncthreads();

  const int e = t >> 2;        // edge within chunk
  const int q = t & 3;         // 16-float quarter of D
  const int r = s_rows[e];
  const int c = s_cols[e] + W_;
  const float w = s_vn[e];

  const size_t roff = ((size_t)g * N_ + r) * D_ + q * 16;
  const size_t coff = ((size_t)g * N_ + c) * D_ + q * 16;
  const float4* __restrict__ sc = (const float4*)(fcur + coff);
  const float4* __restrict__ sr = (const float4*)(fcur + roff);
  float* __restrict__ dr = ftgt + roff;
  float* __restrict__ dc = ftgt + coff;

#pragma unroll
  for (int k = 0; k < 4; ++k) {
    float4 fc = sc[k];
    unsafeAtomicAdd(dr + 4 * k + 0, w * fc.x);
    unsafeAtomicAdd(dr + 4 * k + 1, w * fc.y);
    unsafeAtomicAdd(dr + 4 * k + 2, w * fc.z);
    unsafeAtomicAdd(dr + 4 * k + 3, w * fc.w);
    float4 fr = sr[k];
    unsafeAtomicAdd(dc + 4 * k + 0, w * fr.x);
    unsafeAtomicAdd(dc + 4 * k + 1, w * fr.y);
    unsafeAtomicAdd(dc + 4 * k + 2, w * fr.z);
    unsafeAtomicAdd(dc + 4 * k + 3, w * fr.w);
  }
}

// ---------------- fused: f = y*scale; nrm = WMMA diag(F F^T); acc += f/nrm;
//                  writeback f (next layer input); zero other buffer ----------------
// One wave handles a 16-row tile. A-operand layout per CDNA5 ISA (16x4 f32 A):
// lane L holds row M=L%16; VGPR0/1 hold K-pairs {0,1} (lanes 0-15) / {2,3} (lanes 16-31).
// The B operand of F*F^T in that layout is bit-identical to A, so wmma(a,a,c)
// accumulates the Gram matrix; its diagonal gives squared row norms.

__global__ __launch_bounds__(320) void normacc_kernel(float* __restrict__ ftgt,
                                                      float* __restrict__ fother,
                                                      float* __restrict__ out,
                                                      float scale) {
  const int lane = threadIdx.x & 31;
  const int wave = blockIdx.x * (blockDim.x >> 5) + (threadIdx.x >> 5);
  const int TPG = N_ / 16;                    // 9375 tiles per graph (exact)
  if (wave >= G_ * TPG) return;               // wave-uniform guard (EXEC stays full)

  const int g = wave / TPG;
  const int n0 = (wave % TPG) * 16;
  const int m = lane & 15;
  const int hf = lane >> 4;                   // 0: K%4 in {0,1}, 1: {2,3}
  const int n = n0 + m;
  const size_t rowoff = ((size_t)g * N_ + n) * D_;
  float* __restrict__ p = ftgt + rowoff;

  v2f a[16];
#pragma unroll
  for (int j = 0; j < 16; ++j) {
    float2 t2 = *(const float2*)(p + 4 * j + 2 * hf);
    a[j].x = t2.x * scale;
    a[j].y = t2.y * scale;
  }

  v8f cacc = {0.f, 0.f, 0.f, 0.f, 0.f, 0.f, 0.f, 0.f};
#pragma unroll
  for (int j = 0; j < 16; ++j) {
    cacc = __builtin_amdgcn_wmma_f32_16x16x4_f32(false, a[j], false, a[j],
                                                 (short)0, cacc, false, false);
  }

  // Extract diagonal (row-m squared norm): (m,m) lives in cacc[m%8] at lane m (m<8)
  // or lane m+16 (m>=8) per the 16x16 f32 C/D layout.
  float nrm2 = 0.f;
#pragma unroll
  for (int mm = 0; mm < 16; ++mm) {
    int src = (mm < 8) ? mm : (mm + 16);
    float dv = __shfl(cacc[mm & 7], src, 32);
    if (m == mm) nrm2 = dv;
  }
  const float inv = 1.0f / fmaxf(sqrtf(nrm2), 1e-12f);

  float* __restrict__ op;
  if (n < W_) op = out + ((size_t)g * W_ + n) * D_;
  else        op = out + (size_t)G_ * W_ * D_ + ((size_t)g * S_ + (n - W_)) * D_;

#pragma unroll
  for (int j = 0; j < 16; ++j) {
    const int col = 4 * j + 2 * hf;
    *(float2*)(p + col) = make_float2(a[j].x, a[j].y);   // f = y*scale (next layer input)
    float2 o = *(const float2*)(op + col);
    o.x += a[j].x * inv;
    o.y += a[j].y * inv;
    *(float2*)(op + col) = o;                            // acc += f / nrm
  }

  // Zero the other ping-pong buffer's rows for the next layer's scatter target.
  float* __restrict__ zp = fother + rowoff + hf * 32;
  const float4 z = make_float4(0.f, 0.f, 0.f, 0.f);
#pragma unroll
  for (int k = 0; k < 8; ++k) ((float4*)zp)[k] = z;
}

// ---------------- host wrapper ----------------

extern "C" void kernel_launch(void* const* d_in, const int* in_sizes, int n_in,
                              void* d_out, int out_size, void* d_ws, size_t ws_size,
                              hipStream_t stream) {
  (void)in_sizes; (void)n_in; (void)out_size; (void)ws_size;
  const int*   rows = (const int*)d_in[0];
  const int*   cols = (const int*)d_in[1];
  const float* vals = (const float*)d_in[2];
  const float* wf   = (const float*)d_in[3];
  const float* sf   = (const float*)d_in[4];
  float* out = (float*)d_out;

  // Workspace layout (floats): invs[G*N] | vn[G*E] | f0[G*N*D] | f1[G*N*D]
  float* ws   = (float*)d_ws;
  float* invs = ws;
  float* vn   = invs + (size_t)G_ * N_;
  float* f0   = vn + (size_t)G_ * E_;
  float* f1   = f0 + (size_t)G_ * N_ * D_;

  const int GN_BLOCKS = (G_ * N_ + 255) / 256;
  const int GE_BLOCKS = (G_ * E_ + 255) / 256;

  zero_deg_kernel<<<GN_BLOCKS, 256, 0, stream>>>(invs);
  degree_kernel<<<GE_BLOCKS, 256, 0, stream>>>(rows, cols, vals, invs);
  invsqrt_kernel<<<GN_BLOCKS, 256, 0, stream>>>(invs);
  edgeweight_kernel<<<GE_BLOCKS, 256, 0, stream>>>(rows, cols, vals, invs, vn);
  init_feat_kernel<<<(G_ * N_ * D_ / 4) / 256, 256, 0, stream>>>(wf, sf, f0, f1, out);

  float* fc = f0;
  float* ft = f1;
  for (int i = 0; i < 2; ++i) {
    spmm_kernel<<<(G_ * E_) / CHUNK, 256, 0, stream>>>(rows, cols, vn, fc, ft);
    normacc_kernel<<<(G_ * N_ / 16) / 10, 320, 0, stream>>>(ft, fc, out, 1.0f / (float)(i + 2));
    float* tmp = fc; fc = ft; ft = tmp;
  }
}